// PicMix_22634477650426
// MI455X (gfx1250) — compile-verified
//
#include <hip/hip_runtime.h>

// PicMix mask generation for MI455X (gfx1250, wave32).
//
// Output = (mask_a, mask_b) each (8, 12, 512, 512) f32, concatenated flat.
// Channel groups (c/3): 0 -> ones/zeros, 1 -> ABA checkerboard (i%2==j%2),
// 2 -> row parity (i%2), 3 -> col parity (j%2); mask_b = complement of
// mask_a for groups 1..3, zero for group 0.
//
// Pure store-bandwidth workload: 201.3 MB written, 0 bytes read (outputs are
// independent of the input images). Floor at 23.3 TB/s HBM: ~8.6 us.
// Strategy: wave32-coalesced global_store_b128 with non-temporal hints
// (output > 192 MB L2, written exactly once), fully branchless LUT-based
// pattern selection, group derived from blockIdx.y (block-uniform SALU).

typedef float f32x4 __attribute__((ext_vector_type(4)));

__global__ __launch_bounds__(256) void picmix_masks_kernel(
    float* __restrict__ out, long long nMaskElems) {
  // grid.y = 96 planes (b*12 + c); grid.x = 64 blocks per 512x512 plane.
  // Each thread owns 16 consecutive floats; each wave32 owns one full row.
  const int plane = blockIdx.y;                 // b*12 + c
  const int g     = (plane % 12) / 3;           // group 0..3 (block-uniform)
  const int chunk = blockIdx.x * 256 + threadIdx.x;  // 0..16383 within plane
  const int hp    = (chunk >> 5) & 1;           // row parity (wave-uniform)

  // Value at even / odd columns, LUT indexed by g*2 + hp:
  //   ae bits: g0:{1,1} g1:{1,0} g2:{0,1} g3:{0,0} -> 0x27
  //   ao bits: g0:{1,1} g1:{0,1} g2:{0,1} g3:{1,1} -> 0xEB
  const int idx = g * 2 + hp;
  const int ae  = (0x27 >> idx) & 1;
  const int ao  = (0xEB >> idx) & 1;
  const int nz  = (g != 0) ? 1 : 0;             // mask_b is 0 for group 0
  const int be  = nz & (ae ^ 1);
  const int bo  = nz & (ao ^ 1);

  const float fae = (float)ae, fao = (float)ao;
  const float fbe = (float)be, fbo = (float)bo;
  const f32x4 va = {fae, fao, fae, fao};
  const f32x4 vb = {fbe, fbo, fbe, fbo};

  const long long base = (long long)plane * (512LL * 512) +
                         (long long)chunk * 16;
  f32x4* __restrict__ pa = (f32x4*)(out + base);               // mask_a
  f32x4* __restrict__ pb = (f32x4*)(out + nMaskElems + base);  // mask_b
#pragma unroll
  for (int k = 0; k < 4; ++k) {
    __builtin_nontemporal_store(va, pa + k);  // global_store_b128, TH=NT
    __builtin_nontemporal_store(vb, pb + k);
  }
}

extern "C" void kernel_launch(void* const* d_in, const int* in_sizes, int n_in,
                              void* d_out, int out_size, void* d_ws,
                              size_t ws_size, hipStream_t stream) {
  (void)d_in; (void)in_sizes; (void)n_in; (void)d_ws; (void)ws_size;
  (void)out_size;

  const long long nMaskElems = 8LL * 12 * 512 * 512;  // 25,165,824 per mask

  dim3 grid(64, 96, 1);   // 64 blocks/plane x 96 planes = 6144 blocks
  dim3 block(256, 1, 1);  // 8 wave32 per block

  picmix_masks_kernel<<<grid, block, 0, stream>>>((float*)d_out, nMaskElems);
}